// GAT_71803263255086
// MI455X (gfx1250) — compile-verified
//
#include <hip/hip_runtime.h>
#include <hip/hip_bf16.h>

#define HID 128
#define NGRAPH 512
#define NEG_SLOPE 0.2f
#define BN_EPS 1e-5f

typedef __attribute__((ext_vector_type(2))) float v2f;
typedef __attribute__((ext_vector_type(8))) float v8f;

// order-preserving float <-> uint mapping (for atomicMax over signed floats)
__device__ __forceinline__ unsigned f2o(float f) {
    unsigned u = __float_as_uint(f);
    return (u & 0x80000000u) ? ~u : (u | 0x80000000u);
}
__device__ __forceinline__ float o2f(unsigned u) {
    unsigned v = (u & 0x80000000u) ? (u & 0x7fffffffu) : ~u;
    return __uint_as_float(v);
}
__device__ __forceinline__ float lrelu(float a) { return a > 0.0f ? a : NEG_SLOPE * a; }

// ---------------------------------------------------------------------------
// K0: zero a float buffer
// ---------------------------------------------------------------------------
__global__ void gat_zero(float* __restrict__ p, int n) {
    int i = blockIdx.x * blockDim.x + threadIdx.x;
    if (i < n) p[i] = 0.0f;
}

// ---------------------------------------------------------------------------
// K1: H = X @ W  via V_WMMA_F32_16X16X4_F32.
// One wave computes a 16-row x 128-col strip (8 accumulators of 16x16).
// A layout  (16x4 f32): lane L -> M = L%16, K = (L/16)*2 + {0,1}
// B layout  (4x16 f32): lane L -> N = L%16, K = (L/16)*2 + {0,1}
// C/D layout (16x16 f32): VGPR j, lane L -> row = j + 8*(L/16), col = L%16
// ---------------------------------------------------------------------------
__global__ void gat_gemm_wmma(const float* __restrict__ X, const float* __restrict__ W,
                              float* __restrict__ H, int nrows) {
    int wave = (blockIdx.x * blockDim.x + threadIdx.x) >> 5;
    int m0 = wave * 16;
    if (m0 >= nrows) return;                 // wave-uniform: EXEC stays all-ones for WMMA
    int lane = threadIdx.x & 31;
    int lr = lane & 15;
    int lh = lane >> 4;

    v8f acc[8] = {};
    const float* xrow = X + (size_t)(m0 + lr) * HID;

    for (int k = 0; k < HID; k += 4) {
        int ka = k + lh * 2;
        v2f a;
        a.x = xrow[ka];
        a.y = xrow[ka + 1];
#pragma unroll
        for (int n = 0; n < 8; ++n) {
            v2f b;
            b.x = W[ka * HID + n * 16 + lr];
            b.y = W[(ka + 1) * HID + n * 16 + lr];
            acc[n] = __builtin_amdgcn_wmma_f32_16x16x4_f32(
                false, a, false, b, (short)0, acc[n], false, false);
        }
    }
#pragma unroll
    for (int n = 0; n < 8; ++n)
#pragma unroll
        for (int j = 0; j < 8; ++j)
            H[(size_t)(m0 + j + lh * 8) * HID + n * 16 + lr] = acc[n][j];
}

// ---------------------------------------------------------------------------
// K2: per-node attention logits alpha_src/alpha_dst = h . att_{src,dst};
//     seed segment-max with the self-loop logit (ordered-bits).
// One wave per node; lane handles 4 channels (float4).
// ---------------------------------------------------------------------------
__global__ void gat_node_dots(const float* __restrict__ H, const float* __restrict__ att_s,
                              const float* __restrict__ att_d, float* __restrict__ a_s,
                              float* __restrict__ a_d, unsigned* __restrict__ mbits, int n_nodes) {
    int node = (blockIdx.x * blockDim.x + threadIdx.x) >> 5;
    if (node >= n_nodes) return;
    int lane = threadIdx.x & 31;
    float4 hv = ((const float4*)(H + (size_t)node * HID))[lane];
    float4 ws = ((const float4*)att_s)[lane];
    float4 wd = ((const float4*)att_d)[lane];
    float ps = hv.x * ws.x + hv.y * ws.y + hv.z * ws.z + hv.w * ws.w;
    float pd = hv.x * wd.x + hv.y * wd.y + hv.z * wd.z + hv.w * wd.w;
    for (int off = 16; off > 0; off >>= 1) {
        ps += __shfl_down(ps, off, 32);
        pd += __shfl_down(pd, off, 32);
    }
    if (lane == 0) {
        a_s[node] = ps;
        a_d[node] = pd;
        mbits[node] = f2o(lrelu(ps + pd));   // self-loop seeds the max
    }
}

// ---------------------------------------------------------------------------
// K3: edge pass 1 — segment max over destinations (u32 atomicMax, ordered bits)
// ---------------------------------------------------------------------------
__global__ void gat_edge_max(const int* __restrict__ ei, const float* __restrict__ a_s,
                             const float* __restrict__ a_d, unsigned* __restrict__ mbits,
                             int n_edges) {
    int e = blockIdx.x * blockDim.x + threadIdx.x;
    if (e >= n_edges) return;
    int s = ei[e];
    int d = ei[n_edges + e];
    float a = lrelu(a_s[s] + a_d[d]);
    atomicMax(&mbits[d], f2o(a));
}

// ---------------------------------------------------------------------------
// K4: node init — convert max bits->float (in place), seed softmax denominator
//     and accumulator with the self-loop term. One wave per node.
// ---------------------------------------------------------------------------
__global__ void gat_node_init(const float* __restrict__ H, const float* __restrict__ a_s,
                              const float* __restrict__ a_d, float* __restrict__ m,
                              float* __restrict__ s, float* __restrict__ out_acc, int n_nodes) {
    int node = (blockIdx.x * blockDim.x + threadIdx.x) >> 5;
    if (node >= n_nodes) return;
    int lane = threadIdx.x & 31;
    float mf = o2f(((const unsigned*)m)[node]);
    float es = expf(lrelu(a_s[node] + a_d[node]) - mf);
    if (lane == 0) {
        m[node] = mf;
        s[node] = es;
    }
    float4 hv = ((const float4*)(H + (size_t)node * HID))[lane];
    float4 r;
    r.x = es * hv.x; r.y = es * hv.y; r.z = es * hv.z; r.w = es * hv.w;
    ((float4*)(out_acc + (size_t)node * HID))[lane] = r;
}

// ---------------------------------------------------------------------------
// K5: edge pass 2 — accumulate exp-weighted h[src] into out_acc[dst] and the
//     denominator s[dst]. One wave per edge, lane = 4 channels.
// ---------------------------------------------------------------------------
__global__ void gat_edge_scatter(const int* __restrict__ ei, const float* __restrict__ a_s,
                                 const float* __restrict__ a_d, const float* __restrict__ m,
                                 const float* __restrict__ H, float* __restrict__ s,
                                 float* __restrict__ out_acc, int n_edges) {
    int e = (blockIdx.x * blockDim.x + threadIdx.x) >> 5;
    if (e >= n_edges) return;
    int lane = threadIdx.x & 31;
    int src = ei[e];
    int dst = ei[n_edges + e];
    float w = expf(lrelu(a_s[src] + a_d[dst]) - m[dst]);
    if (lane == 0) atomicAdd(&s[dst], w);
    float4 hv = ((const float4*)(H + (size_t)src * HID))[lane];
    float* o = out_acc + (size_t)dst * HID + lane * 4;
    atomicAdd(o + 0, w * hv.x);
    atomicAdd(o + 1, w * hv.y);
    atomicAdd(o + 2, w * hv.z);
    atomicAdd(o + 3, w * hv.w);
}

// ---------------------------------------------------------------------------
// K6: out = elu(out_acc / s + bias) in place; accumulate per-channel BN stats.
// blockDim = 128 (one thread per channel), grid-stride over nodes.
// ---------------------------------------------------------------------------
__global__ void gat_finalize(float* __restrict__ out_acc, const float* __restrict__ s,
                             const float* __restrict__ bias, float* __restrict__ sums,
                             float* __restrict__ sumsq, int n_nodes) {
    int c = threadIdx.x;
    float b = bias[c];
    float ls = 0.0f, lq = 0.0f;
    for (int n = blockIdx.x; n < n_nodes; n += gridDim.x) {
        float inv = 1.0f / s[n];
        float v = out_acc[(size_t)n * HID + c] * inv + b;
        v = v > 0.0f ? v : expm1f(v);
        out_acc[(size_t)n * HID + c] = v;
        ls += v;
        lq += v * v;
    }
    atomicAdd(&sums[c], ls);
    atomicAdd(&sumsq[c], lq);
}

// ---------------------------------------------------------------------------
// K7: BatchNorm apply (in place -> next layer's x) fused with graph pooling.
// ---------------------------------------------------------------------------
__global__ void gat_bn_pool(float* __restrict__ out_acc, const float* __restrict__ sums,
                            const float* __restrict__ sumsq, const float* __restrict__ gamma,
                            const float* __restrict__ beta, const int* __restrict__ batch,
                            float* __restrict__ pooled, int layer, int n_nodes) {
    int idx = blockIdx.x * blockDim.x + threadIdx.x;
    if (idx >= n_nodes * HID) return;
    int c = idx & (HID - 1);
    int n = idx >> 7;
    float invN = 1.0f / (float)n_nodes;
    float mu = sums[c] * invN;
    float var = sumsq[c] * invN - mu * mu;
    float xv = (out_acc[idx] - mu) / sqrtf(var + BN_EPS) * gamma[c] + beta[c];
    out_acc[idx] = xv;
    atomicAdd(&pooled[(size_t)batch[n] * (3 * HID) + layer * HID + c], xv);
}

// ---------------------------------------------------------------------------
extern "C" void kernel_launch(void* const* d_in, const int* in_sizes, int n_in,
                              void* d_out, int out_size, void* d_ws, size_t ws_size,
                              hipStream_t stream) {
    const float* x_in  = (const float*)d_in[0];
    const int*   ei    = (const int*)d_in[1];
    const int*   batch = (const int*)d_in[2];
    const float* W     = (const float*)d_in[3];
    const float* att_s = (const float*)d_in[4];
    const float* att_d = (const float*)d_in[5];
    const float* bias  = (const float*)d_in[6];
    const float* gamma = (const float*)d_in[7];
    const float* beta  = (const float*)d_in[8];

    const int N = in_sizes[0] / HID;   // 100000
    const int E = in_sizes[1] / 2;     // 1600000

    float* pooled = (float*)d_out;                       // [NGRAPH, 3*HID]
    float* xfinal = (float*)d_out + (size_t)NGRAPH * 3 * HID;  // [N, HID]

    char* w = (char*)d_ws;
    float* hbuf = (float*)w;  w += (size_t)N * HID * sizeof(float);
    float* xb   = (float*)w;  w += (size_t)N * HID * sizeof(float);
    float* asrc = (float*)w;  w += (size_t)N * sizeof(float);
    float* adst = (float*)w;  w += (size_t)N * sizeof(float);
    float* mbuf = (float*)w;  w += (size_t)N * sizeof(float);
    float* sbuf = (float*)w;  w += (size_t)N * sizeof(float);
    float* sums = (float*)w;  // 128 sums + 128 sumsq, contiguous
    float* sumsq = sums + HID;

    const int waves_g = (N + 15) / 16;
    dim3 blk(256);
    dim3 grid_gemm((waves_g * 32 + 255) / 256);
    dim3 grid_node((N * 32 + 255) / 256);
    dim3 grid_edge((E + 255) / 256);
    dim3 grid_edgew(((size_t)E * 32 + 255) / 256);
    dim3 grid_elem(((size_t)N * HID + 255) / 256);

    // zero the pooled output region once per call
    gat_zero<<<(NGRAPH * 3 * HID + 255) / 256, blk, 0, stream>>>(pooled, NGRAPH * 3 * HID);

    for (int l = 0; l < 3; ++l) {
        const float* X  = (l == 0) ? x_in : xb;
        float* oacc     = (l == 2) ? xfinal : xb;   // layer2 writes final x straight to d_out
        const float* Wl = W + (size_t)l * HID * HID;

        gat_gemm_wmma<<<grid_gemm, blk, 0, stream>>>(X, Wl, hbuf, N);
        gat_node_dots<<<grid_node, blk, 0, stream>>>(hbuf, att_s + l * HID, att_d + l * HID,
                                                     asrc, adst, (unsigned*)mbuf, N);
        gat_edge_max<<<grid_edge, blk, 0, stream>>>(ei, asrc, adst, (unsigned*)mbuf, E);
        gat_node_init<<<grid_node, blk, 0, stream>>>(hbuf, asrc, adst, mbuf, sbuf, oacc, N);
        gat_edge_scatter<<<grid_edgew, blk, 0, stream>>>(ei, asrc, adst, mbuf, hbuf, sbuf, oacc, E);
        gat_zero<<<1, 256, 0, stream>>>(sums, 2 * HID);
        gat_finalize<<<512, 128, 0, stream>>>(oacc, sbuf, bias + l * HID, sums, sumsq, N);
        gat_bn_pool<<<grid_elem, blk, 0, stream>>>(oacc, sums, sumsq, gamma + l * HID,
                                                   beta + l * HID, batch, pooled, l, N);
    }
}